// CDAM_1589137900030
// MI455X (gfx1250) — compile-verified
//
#include <hip/hip_runtime.h>
#include <hip/hip_bf16.h>

// ---------------------------------------------------------------------------
// Adaptive local-contrast modulation, fused for MI455X (gfx1250).
//   Pass A: lumi = conv3x3(x, w1) + b1    -- implicit GEMM on fp32 WMMA
//   Pass B: out  = x * (std3x3(lumi)/(mean3x3(lumi)+eps) * w2 + b2)
// Pass B is the memory-bound pass (~268 MB); NT hints keep L2 clean.
// ---------------------------------------------------------------------------

typedef __attribute__((ext_vector_type(2))) float v2f;
typedef __attribute__((ext_vector_type(8))) float v8f;

#define B_ 8
#define C_ 64
#define H_ 256
#define W_ 256
#define EPS_ 1e-5f

// LDS input staging: 8 channels x 18 rows x 68 cols (cols x0wg-1 .. x0wg+66)
#define CCH   8
#define LROWS 18
#define LW    68
#define LPLANE (LROWS * LW)          // 1224
#define LTOTAL (CCH * LPLANE)        // 9792 floats

// ---------------------------------------------------------------------------
// Kernel A: conv1 (3x3, 64->1, pad 1) via V_WMMA_F32_16X16X4_F32.
// Grid: (W/64, H/16, B), block: 128 threads (4 waves, one 16x16 tile each).
// ---------------------------------------------------------------------------
__global__ __launch_bounds__(128)
void cdam_conv_wmma(const float* __restrict__ xin,
                    const float* __restrict__ w1g,
                    const float* __restrict__ b1g,
                    float* __restrict__ lumi)
{
    __shared__ float x_lds[LTOTAL];
    __shared__ float w_lds[C_ * 9];

    const int tid  = threadIdx.x;
    const int lane = tid & 31;
    const int wave = tid >> 5;          // 0..3 -> which 16-wide column tile
    const int ln   = lane & 15;         // A: row M   /  B: col N
    const int g    = lane >> 4;         // K-pair selector (0 or 1)

    const int b   = blockIdx.z;
    const int y0  = blockIdx.y * 16;    // first output row of this block
    const int xwg = blockIdx.x * 64;    // first output col of this block
    const int x0w = xwg + wave * 16;    // first output col of this wave's tile

    // Preload conv weights (64 ch * 3 * 3) into LDS.
    for (int i = tid; i < C_ * 9; i += 128) w_lds[i] = w1g[i];

    v8f acc = {0.f, 0.f, 0.f, 0.f, 0.f, 0.f, 0.f, 0.f};

    for (int cb = 0; cb < C_; cb += CCH) {
        __syncthreads();   // previous chunk's compute done before overwrite

        // Stage 8 channels of x with conv halo (rows y0-1..y0+16, cols xwg-1..xwg+66),
        // zero-filled outside the image.
        for (int i = tid; i < LTOTAL; i += 128) {
            const int c   = i / LPLANE;
            const int rem = i - c * LPLANE;
            const int r   = rem / LW;
            const int col = rem - r * LW;
            const int gy  = y0 - 1 + r;
            const int gx  = xwg - 1 + col;
            float v = 0.f;
            if (gy >= 0 && gy < H_ && gx >= 0 && gx < W_) {
                v = xin[(((size_t)b * C_ + (cb + c)) * H_ + gy) * W_ + gx];
            }
            x_lds[i] = v;
        }
        __syncthreads();

        // Prefetch the next channel chunk while this one is consumed by WMMAs
        // (global_prefetch_b8; covers ~32KB of the 39KB chunk, one line/lane).
        if (cb + CCH < C_) {
            const float* nxt =
                &xin[(((size_t)b * C_ + (cb + CCH)) * H_ + (y0 > 0 ? y0 - 1 : 0)) * W_];
            __builtin_prefetch(nxt + (size_t)tid * 64, 0, 0);
        }

        // Banded implicit GEMM: 3 row-taps x 5 K-chunks (x' = 0..19) x 8 channels.
        for (int dy = 0; dy < 3; ++dy) {
            const int rbase = (ln + dy) * LW;      // LDS row for this lane's A row
            for (int xb = 0; xb <= 16; xb += 4) {
                // Per-lane band offsets for the two K slots this lane carries.
                const int dx0 = xb + 2 * g - ln;   // weight tap index for K=2g
                const int dx1 = dx0 + 1;           // weight tap index for K=2g+1
                const bool a00 = (dx0 == 0), a01 = (dx0 == 1), a02 = (dx0 == 2);
                const bool a10 = (dx1 == 0), a11 = (dx1 == 1), a12 = (dx1 == 2);
                const int acol = xb + 2 * g;       // LDS col of this lane's A pair

#pragma unroll
                for (int c = 0; c < CCH; ++c) {
                    const float* wp = &w_lds[(cb + c) * 9 + dy * 3];
                    const float w0 = wp[0], w1v = wp[1], w2v = wp[2];

                    const float bx = a00 ? w0 : (a01 ? w1v : (a02 ? w2v : 0.f));
                    const float by = a10 ? w0 : (a11 ? w1v : (a12 ? w2v : 0.f));
                    v2f bf; bf.x = bx; bf.y = by;

                    const v2f af = *(const v2f*)(x_lds + c * LPLANE + rbase + acol);

                    acc = __builtin_amdgcn_wmma_f32_16x16x4_f32(
                        /*neg_a=*/false, af, /*neg_b=*/false, bf,
                        /*c_mod=*/(short)0, acc,
                        /*reuse_a=*/false, /*reuse_b=*/false);
                }
            }
        }
    }

    // Write the 16x16 tile: lane holds (M = j + 8*g, N = ln) in acc[j].
    const float b1v = b1g[0];
#pragma unroll
    for (int j = 0; j < 8; ++j) {
        const int row = y0 + j + 8 * g;
        const int col = x0w + ln;
        lumi[((size_t)b * H_ + row) * W_ + col] = acc[j] + b1v;
    }
}

// ---------------------------------------------------------------------------
// Kernel B: 3x3 zero-padded window stats on lumi -> factor; out = x * factor.
// One block per (b, y) row; 256 threads == 256 columns -> fully coalesced.
// x loads are last-use (NT), out stores are streaming (NT) to keep L2 clean.
// ---------------------------------------------------------------------------
__global__ __launch_bounds__(256)
void cdam_contrast_apply(const float* __restrict__ xin,
                         const float* __restrict__ lumi,
                         const float* __restrict__ w2g,
                         const float* __restrict__ b2g,
                         float* __restrict__ out)
{
    const int p  = blockIdx.x * 256 + threadIdx.x;   // flat (b, y, x)
    const int xc = p & (W_ - 1);
    const int y  = (p / W_) & (H_ - 1);
    const int b  = p / (W_ * H_);

    const float* lrow = lumi + (size_t)b * H_ * W_;

    float s = 0.f, s2 = 0.f;
#pragma unroll
    for (int dy = -1; dy <= 1; ++dy) {
#pragma unroll
        for (int dx = -1; dx <= 1; ++dx) {
            const int yy = y + dy;
            const int xx = xc + dx;
            float v = 0.f;
            if (yy >= 0 && yy < H_ && xx >= 0 && xx < W_)
                v = lrow[yy * W_ + xx];
            s  += v;
            s2 = fmaf(v, v, s2);
        }
    }
    const float mean = s * (1.f / 9.f);
    const float ss   = fmaxf(s2 - s * s * (1.f / 9.f), 0.f);   // sum of sq dev
    const float stdv = sqrtf(ss * (1.f / 8.f));                // ddof = 1 (K-1)
    const float contrast = stdv / (mean + EPS_);
    const float factor   = fmaf(contrast, w2g[0], b2g[0]);

    const size_t base = ((size_t)b * C_) * H_ * W_ + (size_t)y * W_ + xc;
#pragma unroll 4
    for (int c = 0; c < C_; ++c) {
        const size_t idx = base + (size_t)c * H_ * W_;
        const float xv = __builtin_nontemporal_load(&xin[idx]);
        __builtin_nontemporal_store(xv * factor, &out[idx]);
    }
}

// ---------------------------------------------------------------------------
extern "C" void kernel_launch(void* const* d_in, const int* in_sizes, int n_in,
                              void* d_out, int out_size, void* d_ws, size_t ws_size,
                              hipStream_t stream)
{
    const float* x  = (const float*)d_in[0];
    const float* w1 = (const float*)d_in[1];
    const float* b1 = (const float*)d_in[2];
    const float* w2 = (const float*)d_in[3];
    const float* b2 = (const float*)d_in[4];
    float* out  = (float*)d_out;
    float* lumi = (float*)d_ws;           // B*H*W floats = 2 MB scratch

    dim3 gridA(W_ / 64, H_ / 16, B_);     // (4, 16, 8)
    cdam_conv_wmma<<<gridA, 128, 0, stream>>>(x, w1, b1, lumi);

    const int npix = B_ * H_ * W_;        // 524288
    cdam_contrast_apply<<<npix / 256, 256, 0, stream>>>(x, lumi, w2, b2, out);
}